// LazyGraphSNN_54589034332195
// MI455X (gfx1250) — compile-verified
//
#include <hip/hip_runtime.h>
#include <cmath>
#include <stdint.h>

// LazyGraphSNN on MI455X (gfx1250).
// Scatter/atomic-bound workload: <=16 MFLOP total vs ~100MB streamed state.
// WMMA intentionally unused (dense reformulation would need a 63GB matrix and
// ~1.3TB of HBM traffic). CDNA5 paths used: global_atomic_add_f32 scatter,
// global_prefetch_b8 row prefetch, wave32 256-thread blocks.
//
// Potentials stored rescaled: q[n] = pot_true(t) * e^{+t/20}. This makes the
// reference's lazy per-target decay (tmask / lu bookkeeping) algebraically
// exact with zero extra kernels: add c*w*e^{t/20}, test q*e^{-t/20} >= 0.3,
// final output = q*e^{-maxT/20}.

#define NI   2048
#define NH   250000
#define NO   32
#define NT   (NI + NH + NO)     // 252080
#define OUTS (NI + NH)          // 252048
#define BLK  256

__global__ void k_init(const unsigned char* __restrict__ spk,
                       float* __restrict__ q, float* __restrict__ cur,
                       int* __restrict__ hf, int* __restrict__ times) {
    int n = blockIdx.x * blockDim.x + threadIdx.x;
    if (n >= NT) return;
    q[n] = 0.0f;
    if (n < NI) {
        int s = spk[n] ? 1 : 0;         // jnp bool -> 1 byte/elem
        hf[n]  = s;
        cur[n] = s ? 2.0f : 0.0f;       // input drive = 2.0 * spike
    } else {
        hf[n]  = 0;
        cur[n] = 0.0f;
    }
    if (n < NO) times[n] = -1;
}

// Phase A (step t): every neuron with cur > 0.1 scatters cur*w into its 32
// targets' rescaled potentials. scale = e^{t/20} (host-computed per step).
__global__ void k_scatter(const float* __restrict__ cur,
                          const float* __restrict__ w,
                          const int*   __restrict__ tgt,
                          float*       __restrict__ q,
                          float scale, int t,
                          const int* __restrict__ maxT, int limit) {
    int i = blockIdx.x * blockDim.x + threadIdx.x;
    if (i >= limit) return;
    if (t >= *maxT) return;             // uniform scalar load
    float c = cur[i];
    if (c <= 0.1f) return;              // only spiking sources do work
    const int4*   tr = reinterpret_cast<const int4*>(tgt + (size_t)i * 32);
    const float4* wr = reinterpret_cast<const float4*>(w  + (size_t)i * 32);
    __builtin_prefetch(wr, 0, 0);       // global_prefetch_b8: pull weight row
    __builtin_prefetch(tr, 0, 0);       // alongside target row (2 cachelines)
    float cs = c * scale;
#pragma unroll
    for (int k = 0; k < 8; ++k) {       // 32 edges as 4-wide vector loads
        int4   tg = tr[k];
        float4 wv = wr[k];
        atomicAdd(q + tg.x, cs * wv.x); // global_atomic_add_f32, no-return
        atomicAdd(q + tg.y, cs * wv.y);
        atomicAdd(q + tg.z, cs * wv.z);
        atomicAdd(q + tg.w, cs * wv.w);
    }
}

// Phase B (step t): threshold, fire-once bookkeeping, output spike times,
// hidden reset, cur <- sp. dec = e^{-t/20}.
__global__ void k_fire(float* __restrict__ q, float* __restrict__ cur,
                       int* __restrict__ hf, int* __restrict__ times,
                       float dec, int t, const int* __restrict__ maxT) {
    int n = blockIdx.x * blockDim.x + threadIdx.x;
    if (n >= NT) return;
    if (t >= *maxT) return;
    float p  = q[n] * dec;              // true potential at time t
    bool  sp = (p >= 0.3f) && (hf[n] == 0);
    cur[n] = sp ? 1.0f : 0.0f;
    if (sp) {
        hf[n] = 1;
        if (n >= OUTS) {
            int k = n - OUTS;           // exactly one thread per output slot
            if (times[k] < 0) times[k] = t;
        } else {
            q[n] = 0.0f;                // hidden reset; outputs keep potential
        }
    }
}

// d_out layout (reference tuple flattened): [0,32) times, [32,64) output
// potentials, [64, 64+NT) fired flags — all as float.
__global__ void k_final(const float* __restrict__ q, const int* __restrict__ hf,
                        const int* __restrict__ times,
                        const int* __restrict__ maxT,
                        float* __restrict__ out) {
    int n = blockIdx.x * blockDim.x + threadIdx.x;
    if (n >= NT) return;
    out[64 + n] = hf[n] ? 1.0f : 0.0f;
    if (n < NO) out[n] = (float)times[n];
    if (n >= OUTS) {
        float dec = expf(-(float)(*maxT) * 0.05f);   // e^{-maxT/TAU}
        out[32 + (n - OUTS)] = q[n] * dec;
    }
}

extern "C" void kernel_launch(void* const* d_in, const int* in_sizes, int n_in,
                              void* d_out, int out_size, void* d_ws, size_t ws_size,
                              hipStream_t stream) {
    (void)in_sizes; (void)n_in; (void)out_size; (void)ws_size;
    const unsigned char* spk  = (const unsigned char*)d_in[0]; // bool[2048]
    const float*         w    = (const float*)d_in[1];         // [NT,32]
    const int*           tgt  = (const int*)d_in[2];           // [NT,32]
    const int*           maxT = (const int*)d_in[3];           // scalar on device

    char* p = (char*)d_ws;
    float* q     = (float*)p; p += (size_t)NT * 4;
    float* cur   = (float*)p; p += (size_t)NT * 4;
    int*   hf    = (int*)p;   p += (size_t)NT * 4;
    int*   times = (int*)p;                       // 32 ints (ws ~3 MB total)

    const int gN = (NT + BLK - 1) / BLK;          // 985 blocks
    const int gI = (NI + BLK - 1) / BLK;

    k_init<<<gN, BLK, 0, stream>>>(spk, q, cur, hf, times);
    for (int t = 0; t < 20; ++t) {                // setup fixes max_timesteps=20;
        float scale = expf((float)t * 0.05f);     // kernels guard t < *maxT
        float dec   = expf(-(float)t * 0.05f);
        int limit = (t == 0) ? NI : NT;           // step 0: only inputs can spike
        int g     = (t == 0) ? gI : gN;
        k_scatter<<<g,  BLK, 0, stream>>>(cur, w, tgt, q, scale, t, maxT, limit);
        k_fire   <<<gN, BLK, 0, stream>>>(q, cur, hf, times, dec, t, maxT);
    }
    k_final<<<gN, BLK, 0, stream>>>(q, hf, times, maxT, (float*)d_out);
}